// RoDAlignAvg_4080218931273
// MI455X (gfx1250) — compile-verified
//
#include <hip/hip_runtime.h>
#include <hip/hip_bf16.h>

// ---------------------------------------------------------------------------
// RoD-align + 2x2 avg pool, MI455X (gfx1250).
//
// The 9x9 bilinear sample grid is ROI-independent; only the inside-box zero
// mask depends on the ROI:
//   Kernel 1: G[b][81][512] bilinear samples per batch -> d_ws (0.66 MB).
//   Kernel 2: per ROI, mask + 2x2 pool G -> 128 MB output (store-bound,
//             ~5.5 us floor at 23.3 TB/s).
//
// CDNA5 paths: global_load_async_to_lds_b128 (ASYNCcnt) double-buffered with
// s_wait_asynccnt 6 pipelining; non-temporal b128 output stores keep the
// L2-resident G slab from being evicted by the 128 MB output stream.
// ---------------------------------------------------------------------------

#define BATCH   4
#define CHANS   512
#define FH      64
#define FW      64
#define NROI    1024
#define GP      81            // 9x9 grid points
#define GRIDN   9
#define CH      64            // channels per LDS chunk
#define NCHUNK  (CHANS / CH)  // 8
#define LSTRIDE 68            // LDS row stride in floats (272 B = 17 * 16 B)
#define WTRANS  162           // float4 transfers per wave per chunk (1296/8)

typedef float v4f __attribute__((ext_vector_type(4)));

// ---------------------------------------------------------------------------
// Kernel 1: bilinear-sample the fixed 9x9 grid for every (batch, channel).
// ---------------------------------------------------------------------------
__global__ __launch_bounds__(256) void rod_precompute(
    const float* __restrict__ feat,   // [B, C, H, W]
    float* __restrict__ G)            // [B, GP, C]
{
    const int n = blockIdx.x * 256 + threadIdx.x;
    if (n >= BATCH * GP * CHANS) return;

    const int c  = n & (CHANS - 1);
    const int gp = (n >> 9) % GP;
    const int b  = n / (GP * CHANS);
    const int gi = gp / GRIDN;
    const int gj = gp % GRIDN;

    const float step = 63.0f / 8.0f;        // (H-1)/(ah-1), exact in fp32
    const float hs = (float)gi * step;
    const float ws = (float)gj * step;
    int h0 = (int)hs; if (h0 > FH - 2) h0 = FH - 2;
    int w0 = (int)ws; if (w0 > FW - 2) w0 = FW - 2;
    const float hr = hs - (float)h0;
    const float wr = ws - (float)w0;

    const float* fp = feat + (((size_t)b * CHANS + c) * FH + h0) * FW + w0;
    const float ul = fp[0];
    const float ur = fp[1];
    const float dl = fp[FW];
    const float dr = fp[FW + 1];

    const float top = ul + wr * (ur - ul);
    const float bot = dl + wr * (dr - dl);
    G[n] = top + hr * (bot - top);
}

// ---------------------------------------------------------------------------
// Wave-uniform async stage of one 81 x 64-channel slab of G into an LDS
// buffer. Each wave issues exactly 6 async instructions (5 full + 1 with two
// lanes), so every wave's ASYNCcnt rises by exactly 6 per chunk -> enables
// s_wait_asynccnt 6 double-buffer pipelining.
// ---------------------------------------------------------------------------
__device__ __forceinline__ void issue_chunk_async(
    unsigned long long gbase, unsigned lds_buf, int chunk, int t)
{
    const int lane = t & 31;
    const int base = (t >> 5) * WTRANS;     // this wave's first transfer
#pragma unroll
    for (int it = 0; it < 5; ++it) {
        const int idx = base + it * 32 + lane;          // < base + 160
        const int gp = idx >> 4;
        const int k  = idx & 15;
        const unsigned long long ga =
            gbase + (unsigned long long)((gp * CHANS + chunk * CH + k * 4) * 4);
        const unsigned lo = lds_buf + (unsigned)((gp * LSTRIDE + k * 4) * 4);
        asm volatile("global_load_async_to_lds_b128 %0, %1, off"
                     :: "v"(lo), "v"(ga) : "memory");
    }
    // transfers base+160, base+161 (lanes 0..1) -- still one instr per wave
    if (lane < 2) {
        const int idx = base + 160 + lane;
        const int gp = idx >> 4;
        const int k  = idx & 15;
        const unsigned long long ga =
            gbase + (unsigned long long)((gp * CHANS + chunk * CH + k * 4) * 4);
        const unsigned lo = lds_buf + (unsigned)((gp * LSTRIDE + k * 4) * 4);
        asm volatile("global_load_async_to_lds_b128 %0, %1, off"
                     :: "v"(lo), "v"(ga) : "memory");
    }
}

// ---------------------------------------------------------------------------
// Kernel 2: one block (8 waves) per ROI.
// ---------------------------------------------------------------------------
__global__ __launch_bounds__(256) void rod_main(
    const float* __restrict__ G,      // [B, GP, C]
    const float* __restrict__ rois,   // [R, 5]
    float* __restrict__ out)          // [R, C, 8, 8]
{
    __shared__ __align__(16) float sG[2][GP * LSTRIDE];
    __shared__ float sm[GP];

    const int r = blockIdx.x;
    const int t = threadIdx.x;

    const float SC   = 1.0f / 16.0f;
    const float step = 63.0f / 8.0f;

    const int b = (int)rois[(size_t)r * 5 + 0];

    // ROI-inside mask over the 9x9 grid; fold the 0.25 pool scale in
    // (exact power-of-two, numerically equivalent).
    if (t < GP) {
        const int gi = t / GRIDN, gj = t % GRIDN;
        const float hs = (float)gi * step;
        const float ws = (float)gj * step;
        const float x1 = rois[(size_t)r * 5 + 1] * SC;
        const float y1 = rois[(size_t)r * 5 + 2] * SC;
        const float x2 = rois[(size_t)r * 5 + 3] * SC;
        const float y2 = rois[(size_t)r * 5 + 4] * SC;
        const bool inside = (hs >= y1) && (hs <= y2) && (ws >= x1) && (ws <= x2);
        sm[t] = inside ? 0.0f : 0.25f;
    }

    const unsigned long long gbase =
        (unsigned long long)(const void*)(G + (size_t)b * GP * CHANS);
    const unsigned lds0 = (unsigned)(unsigned long long)(const void*)&sG[0][0];
    const unsigned lds1 = (unsigned)(unsigned long long)(const void*)&sG[1][0];

    // Prologue: start chunk 0 into buffer 0 while the mask settles.
    issue_chunk_async(gbase, lds0, 0, t);
    __syncthreads();                       // sm visible to all waves

    // Fixed per-thread output cell: float4 covering (i, jb..jb+3).
    const int cell4 = t & 15;
    const int i     = cell4 >> 1;
    const int jb    = (cell4 & 1) * 4;
    const int crow  = t >> 4;              // 0..15 channel row within a pass

    float ma[5], mb[5];
#pragma unroll
    for (int q = 0; q < 5; ++q) {
        ma[q] = sm[(i    ) * GRIDN + jb + q];
        mb[q] = sm[(i + 1) * GRIDN + jb + q];
    }

    float* outr = out + (size_t)r * CHANS * 64;

#pragma unroll 1
    for (int chunk = 0; chunk < NCHUNK; ++chunk) {
        const int buf = chunk & 1;

        if (chunk + 1 < NCHUNK) {
            // Stream next slab into the shadow buffer while we compute.
            issue_chunk_async(gbase, buf ? lds0 : lds1, chunk + 1, t);
            // In-order async completion: <=6 outstanding => chunk's 6 done.
            asm volatile("s_wait_asynccnt 6" ::: "memory");
        } else {
            asm volatile("s_wait_asynccnt 0" ::: "memory");
        }
        __syncthreads();                   // every wave's copies are complete

        const float* sgb = &sG[buf][0];
#pragma unroll
        for (int p = 0; p < 4; ++p) {
            const int cl = crow + p * 16;                 // channel in chunk
            const float* gA = &sgb[((i    ) * GRIDN + jb) * LSTRIDE + cl];
            const float* gB = &sgb[((i + 1) * GRIDN + jb) * LSTRIDE + cl];
            float cs[5];
#pragma unroll
            for (int q = 0; q < 5; ++q)
                cs[q] = ma[q] * gA[q * LSTRIDE] + mb[q] * gB[q * LSTRIDE];

            v4f v;
            v.x = cs[0] + cs[1];
            v.y = cs[1] + cs[2];
            v.z = cs[2] + cs[3];
            v.w = cs[3] + cs[4];

            const int c = chunk * CH + cl;
            // 128 MB output stream, never re-read: NT store keeps G in L2.
            __builtin_nontemporal_store(v, (v4f*)(outr + (size_t)c * 64 + cell4 * 4));
        }
        __syncthreads();   // all reads of buf^1's predecessor done before reuse
    }
}

// ---------------------------------------------------------------------------
extern "C" void kernel_launch(void* const* d_in, const int* in_sizes, int n_in,
                              void* d_out, int out_size, void* d_ws, size_t ws_size,
                              hipStream_t stream) {
    const float* features = (const float*)d_in[0];   // [4,512,64,64] f32
    const float* rois     = (const float*)d_in[1];   // [1024,5] f32
    float* G   = (float*)d_ws;                       // 4*81*512 floats = 0.66 MB
    float* out = (float*)d_out;                      // [1024,512,8,8] f32

    const int total = BATCH * GP * CHANS;
    rod_precompute<<<(total + 255) / 256, 256, 0, stream>>>(features, G);
    rod_main<<<NROI, 256, 0, stream>>>(G, rois, out);
}